// InvariantPolynomial_62775241999220
// MI455X (gfx1250) — compile-verified
//
#include <hip/hip_runtime.h>
#include <hip/hip_bf16.h>

typedef __attribute__((ext_vector_type(2))) float v2f;
typedef __attribute__((ext_vector_type(8))) float v8f;

#define NBLK 512
#define TPB  256
#define WAVES_PER_BLOCK (TPB / 32)

// normalize2mom constant for silu, 1/sqrt(E[silu(z)^2]) (analytic quadrature)
#define SILU_CST 1.6765412f

// fast silu: x * rcp(1+exp(-x)); denominator >= 1 so v_rcp_f32 (~1 ulp) is safe
__device__ __forceinline__ float fast_silu(float x) {
    return x * __builtin_amdgcn_rcpf(1.0f + __expf(-x));
}

__global__ __launch_bounds__(TPB) void invpoly_main(
    const float* __restrict__ pos,      // [E,3]
    const float* __restrict__ x_right,  // [E,1]
    const float* __restrict__ x_left,   // [N,1]
    const float* __restrict__ W1,       // [20,30]
    const float* __restrict__ W2,       // [30,5]
    const int*   __restrict__ origin,   // [E]
    float* __restrict__ block_out,      // [NBLK]
    int E)
{
    __shared__ float lds_w1[20 * 32];   // W1/(sqrt(20)*1.12), N padded 30->32
    __shared__ float lds_w2v[32];       // SILU_CST/sqrt(30) * row-sums of W2
    __shared__ float red[TPB];

    const int tid = threadIdx.x;

    const float w1scale = 1.0f / (sqrtf(20.0f) * 1.12f);
    for (int i = tid; i < 20 * 32; i += TPB) {
        int k = i >> 5, n = i & 31;
        lds_w1[i] = (n < 30) ? W1[k * 30 + n] * w1scale : 0.0f;
    }
    if (tid < 32) {
        float v = 0.0f;
        if (tid < 30) {
            #pragma unroll
            for (int j = 0; j < 5; ++j) v += W2[tid * 5 + j];
        }
        lds_w2v[tid] = v * (SILU_CST / sqrtf(30.0f));
    }
    __syncthreads();

    const int lane = tid & 31;
    const int nlo  = lane & 15;
    const int koff = (lane & 16) ? 2 : 0;   // A/B K-pair selector per wave half

    // Hoist B-matrix fragments (constant W1 tiles) out of the edge loop.
    // B 4x16 layout: VGPR0 = K=(0|2) row, VGPR1 = K=(1|3) row, N = lane&15.
    v2f b0[5], b1[5];
    #pragma unroll
    for (int kc = 0; kc < 5; ++kc) {
        int k = 4 * kc + koff;
        b0[kc].x = lds_w1[k * 32 + nlo];
        b0[kc].y = lds_w1[(k + 1) * 32 + nlo];
        b1[kc].x = lds_w1[k * 32 + nlo + 16];
        b1[kc].y = lds_w1[(k + 1) * 32 + nlo + 16];
    }
    const float w2a = lds_w2v[nlo];
    const float w2b = lds_w2v[nlo + 16];

    const float step  = 3.5f / 21.0f;
    const float istep = 21.0f / 3.5f;

    float contrib = 0.0f;
    const int waveGlobal = blockIdx.x * WAVES_PER_BLOCK + (tid >> 5);
    const int estep = NBLK * WAVES_PER_BLOCK * 16;

    for (int e0 = waveGlobal * 16; e0 < E; e0 += estep) {
        int   e  = e0 + nlo;                 // both wave halves mirror edges 0..15
        int   ec = (e < E) ? e : (E - 1);
        float px = pos[3 * ec + 0];
        float py = pos[3 * ec + 1];
        float pz = pos[3 * ec + 2];
        float d  = sqrtf(px * px + py * py + pz * pz);

        float s = 0.0f;                      // s=0 kills out-of-range tail edges
        if (e < E) s = x_right[ec] * x_left[origin[ec]];

        // A 16x4 fragments: lane holds emb[K=4kc+koff], emb[K=4kc+koff+1]
        v2f a[5];
        #pragma unroll
        for (int kc = 0; kc < 5; ++kc) {
            int   k  = 4 * kc + koff;
            float u0 = (d - step * (float)(k + 1)) * istep;
            float u1 = (d - step * (float)(k + 2)) * istep;
            a[kc].x = __expf(-u0 * u0);
            a[kc].y = __expf(-u1 * u1);
        }

        // H[16x32] = EMB[16x20] @ W1s[20x32] via 10x v_wmma_f32_16x16x4_f32
        v8f acc0 = {};
        v8f acc1 = {};
        #pragma unroll
        for (int kc = 0; kc < 5; ++kc) {
            acc0 = __builtin_amdgcn_wmma_f32_16x16x4_f32(
                false, a[kc], false, b0[kc], (short)0, acc0, false, false);
            acc1 = __builtin_amdgcn_wmma_f32_16x16x4_f32(
                false, a[kc], false, b1[kc], (short)0, acc1, false, false);
        }

        // out += s_M * sum_n silu(H[M,n]) * w2v[n]; lane owns columns nlo, nlo+16
        // rows M = r + 8*(lane>=16); fetch s_M from lane M via shuffle.
        const int srcBase = (lane & 16) >> 1;
        #pragma unroll
        for (int r = 0; r < 8; ++r) {
            float sM = __shfl(s, srcBase + r, 32);
            float t0 = fast_silu(acc0[r]);
            float t1 = fast_silu(acc1[r]);
            contrib += sM * (t0 * w2a + t1 * w2b);
        }
    }

    // deterministic block reduction
    red[tid] = contrib;
    __syncthreads();
    #pragma unroll
    for (int off = TPB / 2; off > 0; off >>= 1) {
        if (tid < off) red[tid] += red[tid + off];
        __syncthreads();
    }
    if (tid == 0) block_out[blockIdx.x] = red[0];
}

__global__ __launch_bounds__(256) void invpoly_reduce(
    const float* __restrict__ part, float* __restrict__ out, int n)
{
    __shared__ float red[256];
    float v = 0.0f;
    for (int i = threadIdx.x; i < n; i += 256) v += part[i];
    red[threadIdx.x] = v;
    __syncthreads();
    #pragma unroll
    for (int off = 128; off > 0; off >>= 1) {
        if (threadIdx.x < off) red[threadIdx.x] += red[threadIdx.x + off];
        __syncthreads();
    }
    if (threadIdx.x == 0) out[0] = red[0];
}

extern "C" void kernel_launch(void* const* d_in, const int* in_sizes, int n_in,
                              void* d_out, int out_size, void* d_ws, size_t ws_size,
                              hipStream_t stream) {
    const float* pos     = (const float*)d_in[0];   // [E,3]
    const float* x_right = (const float*)d_in[1];   // [E,1]
    const float* x_left  = (const float*)d_in[2];   // [N,1]
    const float* W1      = (const float*)d_in[3];   // [20,30]
    const float* W2      = (const float*)d_in[4];   // [30,5]
    const int*   origin  = (const int*)d_in[5];     // [E]
    float*       out     = (float*)d_out;
    float*       partials = (float*)d_ws;           // NBLK floats of scratch

    const int E = in_sizes[0] / 3;

    invpoly_main<<<NBLK, TPB, 0, stream>>>(pos, x_right, x_left, W1, W2,
                                           origin, partials, E);
    invpoly_reduce<<<1, 256, 0, stream>>>(partials, out, NBLK);
}